// ResidualVectorQuantizer_1005022347665
// MI455X (gfx1250) — compile-verified
//
#include <hip/hip_runtime.h>
#include <hip/hip_bf16.h>

// ---------------------------------------------------------------------------
// Residual VQ with Sinkhorn balancing for MI455X (gfx1250, wave32).
//
// - Distance GEMM on the fp32 tensor path (V_WMMA_F32_16X16X4_F32); A-strip
//   staged into LDS with gfx1250 async copies (GLOBAL_LOAD_ASYNC_TO_LDS_B128,
//   ASYNCcnt), since all 8 waves of a block share the same 16 A rows.
// - Sinkhorn in potential form Q = diag(u) K diag(v): K (64MB) stays resident
//   in the 192MB L2; each iteration reads K exactly once (row-normalize fused
//   with the next column-sum), with global_prefetch hints on the row stream.
// - Final row-normalize skipped (argmax-invariant).
// ---------------------------------------------------------------------------

#define B_N   65536
#define D_E   256
#define K_E   256
#define NUM_Q 4
#define SK_ITERS 25
#define SK_EPS 0.003f

typedef __attribute__((ext_vector_type(2))) float v2f;
typedef __attribute__((ext_vector_type(8))) float v8f;
typedef __attribute__((ext_vector_type(4))) int   v4i;

#define AS1 __attribute__((address_space(1)))
#define AS3 __attribute__((address_space(3)))

// Async global->LDS copy (16B), tracked by ASYNCcnt.
__device__ __forceinline__ void async_copy_b128(const float* g, float* l) {
#if __has_builtin(__builtin_amdgcn_global_load_async_to_lds_b128)
    __builtin_amdgcn_global_load_async_to_lds_b128((AS1 v4i*)g, (AS3 v4i*)l, 0, 0);
#else
    unsigned lo = (unsigned)(uintptr_t)(AS3 float*)l;
    asm volatile("global_load_async_to_lds_b128 %0, %1, off"
                 :: "v"(lo), "v"(g) : "memory");
#endif
}

__device__ __forceinline__ void wait_async() {
#if __has_builtin(__builtin_amdgcn_s_wait_asynccnt)
    __builtin_amdgcn_s_wait_asynccnt(0);
#else
    asm volatile("s_wait_asynccnt 0x0" ::: "memory");
#endif
}

// monotonic float <-> uint key for atomic min/max on floats
__device__ __forceinline__ unsigned fkey(float f) {
    unsigned u = __float_as_uint(f);
    return (u & 0x80000000u) ? ~u : (u | 0x80000000u);
}
__device__ __forceinline__ float kinv(unsigned k) {
    unsigned u = (k & 0x80000000u) ? (k ^ 0x80000000u) : ~k;
    return __uint_as_float(u);
}

// ---------------------------------------------------------------------------
// One-time init: residual = x, x_q = 0, loss accumulator = 0.
__global__ void k_init(const float* __restrict__ x, float* __restrict__ res,
                       float* __restrict__ xq, float* __restrict__ loss) {
    size_t i = (size_t)blockIdx.x * blockDim.x + threadIdx.x;
    res[i] = x[i];
    xq[i]  = 0.0f;
    if (i == 0) *loss = 0.0f;
}

// Per-layer init: zero colsum, reset min/max keys.
__global__ void k_layer_init(float* __restrict__ colsum, unsigned* __restrict__ minmax) {
    int t = threadIdx.x;
    colsum[t] = 0.0f;
    if (t == 0) { minmax[0] = 0xFFFFFFFFu; minmax[1] = 0u; }
}

// Row squared-norms: one wave32 per row of a [rows, 256] matrix.
__global__ void k_rownorm(const float* __restrict__ m, float* __restrict__ out) {
    int tid = threadIdx.x, lane = tid & 31, w = tid >> 5;
    int row = blockIdx.x * 8 + w;
    const float* r = m + (size_t)row * 256;
    float s = 0.0f;
#pragma unroll
    for (int j = 0; j < 8; ++j) { float v = r[lane + 32 * j]; s += v * v; }
#pragma unroll
    for (int off = 16; off > 0; off >>= 1) s += __shfl_xor(s, off, 32);
    if (lane == 0) out[row] = s;
}

// ---------------------------------------------------------------------------
// Distance GEMM: d[b,k] = rn[b] + cn[k] - 2 * X[b,:]·Cb[k,:]
// All 8 waves of a block share the same 16 A rows (same tM): stage the
// 16x256 fp32 A strip (16KB) into LDS with async copies, then each wave
// computes one 16x16 tile via V_WMMA_F32_16X16X4_F32 (A from LDS, B global).
// A frag (16x4): lane l<16 -> row l, K={k0,k0+1}; lane>=16 -> K={k0+2,k0+3}
// B frag (4x16): lane l<16 -> col l, K={k0,k0+1}; lane>=16 -> K={k0+2,k0+3}
__global__ void k_gemm_dist(const float* __restrict__ X, const float* __restrict__ Cb,
                            const float* __restrict__ rn, const float* __restrict__ cn,
                            float* __restrict__ dmat, unsigned* __restrict__ minmax) {
    __shared__ float sA[16 * 256];       // 16KB staged A strip
    __shared__ float smin[8], smax[8];
    int tid = threadIdx.x, lane = tid & 31, w = tid >> 5;
    int l16 = lane & 15, half = lane >> 4;
    int tile = blockIdx.x * 8 + w;
    int tM = blockIdx.x >> 1;            // (blockIdx*8)>>4 : shared by all 8 waves
    int tN = tile & 15;
    int rowBase = tM * 16, colBase = tN * 16;

    // Stage A strip: 1024 x float4 chunks spread over 256 threads (ASYNCcnt).
#pragma unroll
    for (int i = 0; i < 4; ++i) {
        int c = tid + 256 * i;
        async_copy_b128(X + (size_t)rowBase * 256 + c * 4, &sA[c * 4]);
    }
    wait_async();
    __syncthreads();

    const float* Ar = &sA[l16 * 256];
    const float* Cr = Cb + (size_t)(colBase + l16) * 256;

    v8f acc = {};
#pragma unroll 4
    for (int k0 = 0; k0 < 256; k0 += 4) {
        float2 a2 = *(const float2*)(Ar + k0 + 2 * half);   // ds_load_b64
        float2 b2 = *(const float2*)(Cr + k0 + 2 * half);   // global_load_b64
        v2f a; a[0] = a2.x; a[1] = a2.y;
        v2f b; b[0] = b2.x; b[1] = b2.y;
        acc = __builtin_amdgcn_wmma_f32_16x16x4_f32(
            false, a, false, b, (short)0, acc, false, false);
    }

    int col = colBase + l16;
    float cnv = cn[col];
    float lmin = 3.4e38f, lmax = -3.4e38f;
#pragma unroll
    for (int j = 0; j < 8; ++j) {
        int row = rowBase + 8 * half + j;          // C/D layout: VGPR j, halves split M
        float val = rn[row] + cnv - 2.0f * acc[j];
        dmat[(size_t)row * 256 + col] = val;
        lmin = fminf(lmin, val);
        lmax = fmaxf(lmax, val);
    }
#pragma unroll
    for (int off = 16; off > 0; off >>= 1) {
        lmin = fminf(lmin, __shfl_xor(lmin, off, 32));
        lmax = fmaxf(lmax, __shfl_xor(lmax, off, 32));
    }
    if (lane == 0) { smin[w] = lmin; smax[w] = lmax; }
    __syncthreads();
    if (tid == 0) {
        float mn = smin[0], mx = smax[0];
        for (int i = 1; i < 8; ++i) { mn = fminf(mn, smin[i]); mx = fmaxf(mx, smax[i]); }
        atomicMin(&minmax[0], fkey(mn));
        atomicMax(&minmax[1], fkey(mx));
    }
}

// ---------------------------------------------------------------------------
// K = exp(-(d - mid)/(amp*eps)) in place; fused first column-sum (u0 == 1).
__global__ void k_exp_colsum(float* __restrict__ dmat, const unsigned* __restrict__ minmax,
                             float* __restrict__ colsum) {
    int t = threadIdx.x;
    float mn = kinv(minmax[0]), mx = kinv(minmax[1]);
    float mid = 0.5f * (mx + mn);
    float amp = mx - mid + 1e-5f;
    float sc  = -1.0f / (amp * SK_EPS);
    size_t base = (size_t)blockIdx.x * 64 * 256;
    float acc = 0.0f;
    for (int r = 0; r < 64; ++r) {
        size_t idx = base + (size_t)r * 256 + t;
        float q = __expf((dmat[idx] - mid) * sc);
        dmat[idx] = q;
        acc += q;
    }
    atomicAdd(&colsum[t], acc);
}

// v[k] = 1/(K * colsum[k]); colsum re-zeroed for the next accumulation.
__global__ void k_vupd(float* __restrict__ v, float* __restrict__ colsum) {
    int t = threadIdx.x;
    v[t] = 1.0f / (256.0f * colsum[t]);
    colsum[t] = 0.0f;
}

// Fused row-normalize + next column-sum: for each row b,
//   u = 1/(B * sum_k K[b,k] v[k]);  colsum'[k] += u * K[b,k]
// One wave per 32 rows; K elements stay in registers between the two uses.
// global_prefetch hints pull the next row while the shuffle reduce runs.
__global__ void k_row_colsum(const float* __restrict__ Km, const float* __restrict__ v,
                             float* __restrict__ colsum) {
    __shared__ float sv[256];
    __shared__ float sc[8][256];
    int t = threadIdx.x, lane = t & 31, w = t >> 5;
    sv[t] = v[t];
    __syncthreads();

    float cacc[8] = {0.f, 0.f, 0.f, 0.f, 0.f, 0.f, 0.f, 0.f};
    int rowBase = blockIdx.x * 256 + w * 32;
    for (int r = 0; r < 32; ++r) {
        const float* Kr = Km + (size_t)(rowBase + r) * 256;
        if (r + 1 < 32)
            __builtin_prefetch(Km + (size_t)(rowBase + r + 1) * 256 + lane * 8, 0, 1);
        float kv[8]; float s = 0.0f;
#pragma unroll
        for (int j = 0; j < 8; ++j) {
            int k = lane + 32 * j;
            float q = Kr[k];
            kv[j] = q;
            s += q * sv[k];
        }
#pragma unroll
        for (int off = 16; off > 0; off >>= 1) s += __shfl_xor(s, off, 32);
        float un = 1.0f / (65536.0f * s);
#pragma unroll
        for (int j = 0; j < 8; ++j) cacc[j] += un * kv[j];
    }
#pragma unroll
    for (int j = 0; j < 8; ++j) sc[w][lane + 32 * j] = cacc[j];
    __syncthreads();
    float tot = 0.0f;
#pragma unroll
    for (int i = 0; i < 8; ++i) tot += sc[i][t];
    atomicAdd(&colsum[t], tot);
}

// ---------------------------------------------------------------------------
// Final per-layer pass: argmax_k K[b,k]*v[k] (row scaling by u is argmax-
// invariant), gather codebook row, accumulate x_q, update residual, loss.
__global__ void k_finalize_layer(const float* __restrict__ Km, const float* __restrict__ v,
                                 const float* __restrict__ Cb, float* __restrict__ res,
                                 float* __restrict__ xq, float* __restrict__ outIdx,
                                 float* __restrict__ loss, int layer) {
    __shared__ float sv[256];
    int t = threadIdx.x, lane = t & 31, w = t >> 5;
    sv[t] = v[t];
    __syncthreads();

    int b = blockIdx.x * 8 + w;
    const float* Kr = Km + (size_t)b * 256;
    float m = -3.4e38f; int mi = 0;
#pragma unroll
    for (int j = 0; j < 8; ++j) {
        int k = lane + 32 * j;
        float q = Kr[k] * sv[k];
        if (q > m) { m = q; mi = k; }
    }
#pragma unroll
    for (int off = 16; off > 0; off >>= 1) {      // (max, lowest index) reduce
        float om = __shfl_xor(m, off, 32);
        int   oi = __shfl_xor(mi, off, 32);
        if (om > m || (om == m && oi < mi)) { m = om; mi = oi; }
    }

    const float* cbr = Cb + (size_t)mi * 256;
    float ss = 0.0f;
#pragma unroll
    for (int j = 0; j < 8; ++j) {
        int k = lane + 32 * j;
        size_t idx = (size_t)b * 256 + k;
        float c = cbr[k];
        float r = res[idx];
        float df = c - r;
        ss += df * df;
        xq[idx] += c;          // exclusive row ownership; layers serialized
        res[idx] = r - c;
    }
#pragma unroll
    for (int off = 16; off > 0; off >>= 1) ss += __shfl_xor(ss, off, 32);
    if (lane == 0) {
        atomicAdd(loss, ss);
        outIdx[(size_t)b * 4 + layer] = (float)mi;
    }
}

// mean over layers of (codebook_loss + commit_loss) = 2 * sum_sq / (NUM_Q*B*D)
__global__ void k_write_loss(const float* __restrict__ loss, float* __restrict__ out) {
    out[0] = loss[0] * (2.0f / (4.0f * (float)B_N * (float)D_E));
}

// ---------------------------------------------------------------------------
extern "C" void kernel_launch(void* const* d_in, const int* in_sizes, int n_in,
                              void* d_out, int out_size, void* d_ws, size_t ws_size,
                              hipStream_t stream) {
    (void)in_sizes; (void)n_in; (void)out_size; (void)ws_size;

    const float* x         = (const float*)d_in[0];   // [B, 256]
    /* d_in[1] = labels (unused by the math) */
    const float* codebooks = (const float*)d_in[2];   // [4, 256, 256]

    float* out     = (float*)d_out;
    float* xq      = out;                              // [B*256]
    float* outLoss = out + (size_t)B_N * D_E;          // [1]
    float* outIdx  = outLoss + 1;                      // [B*4], indices stored as floats

    char* ws = (char*)d_ws;
    float*    dmat    = (float*)ws;  ws += (size_t)B_N * 256 * sizeof(float); // d, then K (64MB, L2-resident)
    float*    res     = (float*)ws;  ws += (size_t)B_N * 256 * sizeof(float); // residual (64MB)
    float*    rn      = (float*)ws;  ws += (size_t)B_N * sizeof(float);       // ||r_b||^2
    float*    cn      = (float*)ws;  ws += 256 * sizeof(float);               // ||c_k||^2
    float*    colsum  = (float*)ws;  ws += 256 * sizeof(float);
    float*    v       = (float*)ws;  ws += 256 * sizeof(float);
    unsigned* minmax  = (unsigned*)ws; ws += 64;                              // 2 keys + pad
    float*    lossAcc = (float*)ws;

    k_init<<<(B_N * 256) / 256, 256, 0, stream>>>(x, res, xq, lossAcc);

    for (int q = 0; q < NUM_Q; ++q) {
        const float* Cb = codebooks + (size_t)q * 256 * 256;

        k_layer_init<<<1, 256, 0, stream>>>(colsum, minmax);
        k_rownorm<<<B_N / 8, 256, 0, stream>>>(res, rn);
        k_rownorm<<<256 / 8, 256, 0, stream>>>(Cb, cn);
        k_gemm_dist<<<(B_N / 16) * (256 / 16) / 8, 256, 0, stream>>>(res, Cb, rn, cn, dmat, minmax);
        k_exp_colsum<<<B_N / 64, 256, 0, stream>>>(dmat, minmax, colsum);

        for (int it = 0; it < SK_ITERS; ++it) {
            k_vupd<<<1, 256, 0, stream>>>(v, colsum);
            if (it < SK_ITERS - 1)   // last row-normalize is argmax-invariant
                k_row_colsum<<<B_N / 256, 256, 0, stream>>>(dmat, v, colsum);
        }

        k_finalize_layer<<<B_N / 8, 256, 0, stream>>>(dmat, v, Cb, res, xq, outIdx, lossAcc, q);
    }

    k_write_loss<<<1, 1, 0, stream>>>(lossAcc, outLoss);
}